// ThinPlateSpline2D_77945066487981
// MI455X (gfx1250) — compile-verified
//
#include <hip/hip_runtime.h>
#include <hip/hip_bf16.h>
#include <math.h>

// ---------------------------------------------------------------------------
// Thin-plate-spline warp, B=16, H=W=224, C=8, 8x8 control grid.
//   Kernel 1: build + solve the 67x67 TPS system for all 16 batches (32 RHS
//             columns) with Gauss-Jordan + partial pivoting. Output WV[68][32]
//             (K padded to 68) into d_ws.
//   Kernel 2: per 16-pixel tile, compute loc = Phi(16x68) @ WV(68x32) with
//             v_wmma_f32_16x16x4_f32 (batches packed into N), then bilinear
//             resample vol with float4 (b128) gathers.
// ---------------------------------------------------------------------------

typedef __attribute__((ext_vector_type(2))) float v2f;
typedef __attribute__((ext_vector_type(8))) float v8f;

#define IMG 224
#define NCP 64          // 8x8 control points
#define NK  67          // 64 + affine(3)
#define NKP 68          // padded K for WMMA (multiple of 4)
#define NB  32          // 16 batches * 2 dims packed into N
#define EPSR2 1e-10f

// ---------------------------------------------------------------------------
// Kernel 1: solve  lhs(67x67) @ WV = RHS(67x32)  -> WV[68][32] in d_ws
// ---------------------------------------------------------------------------
__global__ __launch_bounds__(128) void tps_solve_kernel(
    const float* __restrict__ cp_offsets,   // (16,1,64,2)
    float* __restrict__ wv_out)             // (68,32)
{
    __shared__ float A[NK][NKP];   // 67x68 (padded column)
    __shared__ float R[NK][NB];    // 67x32
    __shared__ int   s_piv;

    const int tid = threadIdx.x;
    const float inv7 = 1.0f / 7.0f;

    // Build lhs = [[phi(cp,cp), Bm],[Bm^T, 0]],  cp_n = ((n>>3)/7, (n&7)/7)
    for (int idx = tid; idx < NK * NK; idx += blockDim.x) {
        int i = idx / NK, j = idx % NK;
        float v;
        if (i < NCP && j < NCP) {
            float dy = (float)((i >> 3) - (j >> 3)) * inv7;
            float dx = (float)((i & 7) - (j & 7)) * inv7;
            float r2 = dy * dy + dx * dx;
            v = 0.5f * r2 * logf(fmaxf(r2, EPSR2));
        } else if (i < NCP) {          // columns 64..66: [cp_y, cp_x, 1]
            v = (j == 64) ? (float)(i >> 3) * inv7
              : (j == 65) ? (float)(i & 7) * inv7 : 1.0f;
        } else if (j < NCP) {          // rows 64..66: Bm^T
            v = (i == 64) ? (float)(j >> 3) * inv7
              : (i == 65) ? (float)(j & 7) * inv7 : 1.0f;
        } else {
            v = 0.0f;
        }
        A[i][j] = v;
    }
    // RHS: rows 0..63 = cp + offset (per batch/dim), rows 64..66 = 0
    for (int idx = tid; idx < NK * NB; idx += blockDim.x) {
        int i = idx / NB, n = idx % NB;
        float v = 0.0f;
        if (i < NCP) {
            int b = n >> 1, d = n & 1;
            float cpc = (d == 0) ? (float)(i >> 3) * inv7
                                 : (float)(i & 7) * inv7;
            v = cpc + cp_offsets[((size_t)b * NCP + i) * 2 + d];
        }
        R[i][n] = v;
    }
    __syncthreads();

    // Gauss-Jordan with partial pivoting
    for (int k = 0; k < NK; ++k) {
        if (tid == 0) {
            int piv = k;
            float best = fabsf(A[k][k]);
            for (int i = k + 1; i < NK; ++i) {
                float a = fabsf(A[i][k]);
                if (a > best) { best = a; piv = i; }
            }
            s_piv = piv;
        }
        __syncthreads();
        const int piv = s_piv;
        if (piv != k) {
            for (int j = tid; j < NK; j += blockDim.x) {
                float t = A[k][j]; A[k][j] = A[piv][j]; A[piv][j] = t;
            }
            for (int j = tid; j < NB; j += blockDim.x) {
                float t = R[k][j]; R[k][j] = R[piv][j]; R[piv][j] = t;
            }
        }
        __syncthreads();
        if (tid < NK && tid != k) {
            const float f = A[tid][k] / A[k][k];
            for (int j = k; j < NK; ++j) A[tid][j] -= f * A[k][j];
            for (int j = 0; j < NB; ++j) R[tid][j] -= f * R[k][j];
        }
        __syncthreads();
    }

    // Normalize diagonal, write WV (pad row 67 with zeros)
    for (int idx = tid; idx < NKP * NB; idx += blockDim.x) {
        int k = idx / NB, n = idx % NB;
        wv_out[idx] = (k < NK) ? (R[k][n] / A[k][k]) : 0.0f;
    }
}

// ---------------------------------------------------------------------------
// Kernel 2: WMMA loc-GEMM + bilinear resample
// ---------------------------------------------------------------------------
__device__ __forceinline__ float phi_col(int k, float qy, float qx) {
    if (k < NCP) {
        float cy = (float)(k >> 3) * (1.0f / 7.0f);
        float cx = (float)(k & 7) * (1.0f / 7.0f);
        float dy = qy - cy, dx = qx - cx;
        float r2 = dy * dy + dx * dx;
        return 0.5f * r2 * logf(fmaxf(r2, EPSR2));
    }
    if (k == 64) return qy;
    if (k == 65) return qx;
    if (k == 66) return 1.0f;
    return 0.0f;   // k == 67 padding
}

__device__ __forceinline__ void corner_gather(const float* __restrict__ vol,
                                              int b, int yi, int xi, float w,
                                              float r[8]) {
    if ((unsigned)xi < (unsigned)IMG && (unsigned)yi < (unsigned)IMG) {
        const float4* vp = (const float4*)(vol +
            ((size_t)((b * IMG + yi) * IMG + xi)) * 8);
        float4 a = vp[0];
        float4 c = vp[1];
        r[0] += a.x * w; r[1] += a.y * w; r[2] += a.z * w; r[3] += a.w * w;
        r[4] += c.x * w; r[5] += c.y * w; r[6] += c.z * w; r[7] += c.w * w;
    }
}

__global__ __launch_bounds__(128) void tps_warp_kernel(
    const float* __restrict__ vol,   // (16,224,224,8)
    const float* __restrict__ wv,    // (68,32)
    float* __restrict__ out)         // (16,224,224,8)
{
    __shared__ float sWV[NKP * NB];        // 68x32 = 8.7 KB
    __shared__ float sLoc[4][16][NB];      // per-wave 16x32 loc tile

    const int tid  = threadIdx.x;
    const int wave = tid >> 5;
    const int lane = tid & 31;

    for (int i = tid; i < NKP * NB; i += 128) sWV[i] = wv[i];
    __syncthreads();

    // One wave handles one 16-pixel tile (all tiles stay within a row:
    // 224 % 16 == 0)
    const int tile  = blockIdx.x * 4 + wave;
    const int m0    = tile * 16;
    const int mrow  = lane & 15;
    const int khalf = lane >> 4;          // 0: K%4 in {0,1}; 1: {2,3}
    const int nb    = lane & 15;          // N column within N-halftile

    const int m  = m0 + mrow;
    const float qy = (float)(m / IMG) * (1.0f / 223.0f);
    const float qx = (float)(m % IMG) * (1.0f / 223.0f);

    v8f acc0 = {0.f,0.f,0.f,0.f,0.f,0.f,0.f,0.f};   // N = 0..15
    v8f acc1 = {0.f,0.f,0.f,0.f,0.f,0.f,0.f,0.f};   // N = 16..31

#pragma unroll
    for (int s = 0; s < NKP / 4; ++s) {
        const int k0 = 4 * s + 2 * khalf;
        v2f a, b0, b1;
        a.x  = phi_col(k0,     qy, qx);
        a.y  = phi_col(k0 + 1, qy, qx);
        b0.x = sWV[(k0    ) * NB + nb];
        b0.y = sWV[(k0 + 1) * NB + nb];
        b1.x = sWV[(k0    ) * NB + 16 + nb];
        b1.y = sWV[(k0 + 1) * NB + 16 + nb];
        acc0 = __builtin_amdgcn_wmma_f32_16x16x4_f32(
                   false, a, false, b0, (short)0, acc0, false, false);
        acc1 = __builtin_amdgcn_wmma_f32_16x16x4_f32(
                   false, a, false, b1, (short)0, acc1, false, false);
    }

    // D layout: VGPR j, lanes 0-15 -> M=j ; lanes 16-31 -> M=8+j ; N=lane&15
#pragma unroll
    for (int j = 0; j < 8; ++j) {
        sLoc[wave][khalf * 8 + j][nb]      = acc0[j];
        sLoc[wave][khalf * 8 + j][16 + nb] = acc1[j];
    }
    __syncthreads();

    // Bilinear resample: 16 pixels x 16 batches = 256 tasks, 8 per lane.
    const int yrow  = m0 / IMG;
    const int xbase = m0 % IMG;
#pragma unroll
    for (int it = 0; it < 8; ++it) {
        const int t = it * 32 + lane;
        const int b = t >> 4;
        const int p = t & 15;
        // loc[...,::-1] * (W-1, H-1): x from dim1, y from dim0
        const float xc = sLoc[wave][p][2 * b + 1] * 223.0f;
        const float yc = sLoc[wave][p][2 * b]     * 223.0f;
        const float x0f = floorf(xc), y0f = floorf(yc);
        const float wx = xc - x0f, wy = yc - y0f;
        const int x0 = (int)x0f, y0 = (int)y0f;

        float r[8] = {0.f,0.f,0.f,0.f,0.f,0.f,0.f,0.f};
        corner_gather(vol, b, y0,     x0,     (1.0f - wx) * (1.0f - wy), r);
        corner_gather(vol, b, y0,     x0 + 1, wx * (1.0f - wy),          r);
        corner_gather(vol, b, y0 + 1, x0,     (1.0f - wx) * wy,          r);
        corner_gather(vol, b, y0 + 1, x0 + 1, wx * wy,                   r);

        float* o = out + ((size_t)((b * IMG + yrow) * IMG + (xbase + p))) * 8;
        ((float4*)o)[0] = make_float4(r[0], r[1], r[2], r[3]);
        ((float4*)o)[1] = make_float4(r[4], r[5], r[6], r[7]);
    }
}

// ---------------------------------------------------------------------------
extern "C" void kernel_launch(void* const* d_in, const int* in_sizes, int n_in,
                              void* d_out, int out_size, void* d_ws, size_t ws_size,
                              hipStream_t stream) {
    const float* vol  = (const float*)d_in[0];   // (16,224,224,8) f32
    const float* coff = (const float*)d_in[1];   // (16,1,64,2)   f32
    float* out = (float*)d_out;                  // (16,224,224,8) f32
    float* wv  = (float*)d_ws;                   // 68*32 f32 scratch

    tps_solve_kernel<<<1, 128, 0, stream>>>(coff, wv);

    // 224*224/16 = 3136 tiles, 4 tiles (waves) per 128-thread block
    tps_warp_kernel<<<3136 / 4, 128, 0, stream>>>(vol, wv, out);
}